// LSTM_79216376807673
// MI455X (gfx1250) — compile-verified
//
#include <hip/hip_runtime.h>

// ---------------------------------------------------------------------------
// 2-layer peephole LSTM forward on gfx1250 (MI455X), fp32, wave32 + WMMA.
//
// Reference quirks faithfully kept:
//   i = sigmoid(gi + Wc[0]*c);  g = tanh(gg);  o = sigmoid(go + Wc[2]*c)
//   f gate is DEAD (computed-but-unused in reference) -> we skip it entirely.
//   ct = g*c + i*g              (forget gate not applied)
//   ht = o*tanh(c_old)          (uses OLD cell state)
//
// Per (layer, timestep): fused kernel computes a 16x16 tile of each live gate
// with chained V_WMMA_F32_16X16X4_F32 over K = d_in + H, then does the
// elementwise update. One wave per gate (3 waves / WG), 256 WGs.
// ---------------------------------------------------------------------------

typedef float v2f __attribute__((ext_vector_type(2)));
typedef float v8f __attribute__((ext_vector_type(8)));

#define H_DIM   1024
#define N_BATCH 64
#define T_STEPS 256
#define IN_DIM  512

__device__ __forceinline__ float sigmoid_f(float x) {
    return 1.0f / (1.0f + __expf(-x));
}
__device__ __forceinline__ float tanh_f(float x) {
    // numerically-guarded fast tanh via exp
    x = fminf(fmaxf(x, -20.0f), 20.0f);
    float e = __expf(2.0f * x);
    return (e - 1.0f) / (e + 1.0f);
}

// gates handled by the 3 waves: wave 0 -> i (row block 0), wave 1 -> g (row
// block 2), wave 2 -> o (row block 3) of the [4H, K] weight matrices.
__device__ __forceinline__ int wave_to_gate(int w) { return (w == 0) ? 0 : (w + 1); }

// WMMA f32 16x16x4 fragment layouts (CDNA5 ISA 7.12.2):
//   A (16x4):  lanes 0-15 : M=lane, VGPR0=K0, VGPR1=K1
//              lanes 16-31: M=lane-16, VGPR0=K2, VGPR1=K3
//   B (4x16):  lanes 0-15 : N=lane, VGPR0=K0 row, VGPR1=K1 row
//              lanes 16-31: N=lane-16, VGPR0=K2 row, VGPR1=K3 row
//   C/D (16x16): VGPR r, lane L -> M = r + 8*(L>>4), N = L&15
// Both A and B fragments are one float2 load per lane (contiguous in K).

__global__ __launch_bounds__(96)
void lstm_step_kernel(const float* __restrict__ xbase, int x_stride, int d_in,
                      const float* __restrict__ Wx,          // [4H, d_in] row-major
                      const float* __restrict__ hbase, int h_stride, // nullptr at t==0
                      const float* __restrict__ Wh,          // [4H, H] row-major
                      const float* __restrict__ b,           // [4H]
                      const float* __restrict__ Wc,          // [3, H]
                      float* __restrict__ c,                 // [N, H] in-place cell state
                      float* __restrict__ hout, int hout_stride)
{
    const int lane = threadIdx.x & 31;
    const int wav  = threadIdx.x >> 5;       // 0..2
    const int gate = wave_to_gate(wav);      // 0 (i), 2 (g), 3 (o)
    const int hc0  = blockIdx.x * 16;        // hidden-column tile base
    const int n0   = blockIdx.y * 16;        // batch-row tile base
    const int half = lane >> 4;              // K-half (0: K0/K1, 1: K2/K3)
    const int l16  = lane & 15;

    const int jrow = gate * H_DIM + hc0 + l16;   // weight row this lane feeds B with

    v8f acc = {};

    // ---- x_t @ Wx^T ------------------------------------------------------
    {
        const float* Ap = xbase + (n0 + l16) * x_stride + 2 * half;
        const float* Bp = Wx + (long)jrow * d_in + 2 * half;
        #pragma unroll 8
        for (int k = 0; k < d_in; k += 4) {
            v2f a = *(const v2f*)(Ap + k);
            v2f w = *(const v2f*)(Bp + k);
            acc = __builtin_amdgcn_wmma_f32_16x16x4_f32(
                      false, a, false, w, (short)0, acc, false, false);
        }
    }
    // ---- h_{t-1} @ Wh^T  (skipped at t==0: h is all zeros) ---------------
    if (hbase != nullptr) {
        const float* Ap = hbase + (n0 + l16) * h_stride + 2 * half;
        const float* Bp = Wh + (long)jrow * H_DIM + 2 * half;
        #pragma unroll 8
        for (int k = 0; k < H_DIM; k += 4) {
            v2f a = *(const v2f*)(Ap + k);
            v2f w = *(const v2f*)(Bp + k);
            acc = __builtin_amdgcn_wmma_f32_16x16x4_f32(
                      false, a, false, w, (short)0, acc, false, false);
        }
    }

    // ---- exchange gate tiles through LDS ---------------------------------
    __shared__ float gsm[3][16][16];   // [wave(gate)][n_local][col]
    const float bias = b[jrow];
    #pragma unroll
    for (int r = 0; r < 8; ++r)
        gsm[wav][r + 8 * half][l16] = acc[r] + bias;
    __syncthreads();

    // ---- elementwise LSTM update (256 elements / 96 threads) -------------
    for (int e = threadIdx.x; e < 256; e += 96) {
        const int nl  = e >> 4;
        const int col = e & 15;
        const int n   = n0 + nl;
        const int j   = hc0 + col;

        const float c_old = c[n * H_DIM + j];
        const float gi = gsm[0][nl][col];
        const float gg = gsm[1][nl][col];
        const float go = gsm[2][nl][col];

        const float iv = sigmoid_f(gi + Wc[0 * H_DIM + j] * c_old);
        const float gv = tanh_f(gg);
        const float ov = sigmoid_f(go + Wc[2 * H_DIM + j] * c_old);

        const float ct = gv * c_old + iv * gv;   // faithful: no forget gate
        const float ht = ov * tanh_f(c_old);     // faithful: old cell state

        c[n * H_DIM + j] = ct;
        hout[n * hout_stride + j] = ht;
    }
}

__global__ void zero_kernel(float* __restrict__ p, int n) {
    int i = blockIdx.x * 256 + threadIdx.x;
    if (i < n) p[i] = 0.0f;
}

// Append (h_last [N,H], c_last [N,H]) after the [N,T,H] sequence output.
__global__ void finalize_kernel(const float* __restrict__ c1, float* __restrict__ out) {
    int i = blockIdx.x * 256 + threadIdx.x;
    if (i < N_BATCH * H_DIM) {
        const int n = i / H_DIM;
        const int j = i - n * H_DIM;
        out[(long)N_BATCH * T_STEPS * H_DIM + i] =
            out[(long)n * T_STEPS * H_DIM + (long)(T_STEPS - 1) * H_DIM + j];
        out[(long)N_BATCH * T_STEPS * H_DIM + (long)N_BATCH * H_DIM + i] = c1[i];
    }
}

extern "C" void kernel_launch(void* const* d_in, const int* in_sizes, int n_in,
                              void* d_out, int out_size, void* d_ws, size_t ws_size,
                              hipStream_t stream) {
    const float* x   = (const float*)d_in[0];
    const float* Wx0 = (const float*)d_in[1];
    const float* Wh0 = (const float*)d_in[2];
    const float* b0  = (const float*)d_in[3];
    const float* Wc0 = (const float*)d_in[4];
    const float* Wx1 = (const float*)d_in[5];
    const float* Wh1 = (const float*)d_in[6];
    const float* b1  = (const float*)d_in[7];
    const float* Wc1 = (const float*)d_in[8];
    float* out = (float*)d_out;
    float* ws  = (float*)d_ws;

    // workspace layout (floats): c0 | c1 | h0 ping | h0 pong   (1 MB total)
    float* c0  = ws;
    float* c1  = ws + N_BATCH * H_DIM;
    float* h0a = ws + 2 * N_BATCH * H_DIM;
    float* h0b = ws + 3 * N_BATCH * H_DIM;

    // cell states start at zero every call (harness does not re-zero ws)
    zero_kernel<<<(2 * N_BATCH * H_DIM + 255) / 256, 256, 0, stream>>>(ws, 2 * N_BATCH * H_DIM);

    const dim3 grid(H_DIM / 16, N_BATCH / 16);   // 64 x 4 workgroups
    const dim3 block(96);                        // 3 wave32 (one per live gate)

    for (int t = 0; t < T_STEPS; ++t) {
        float*       h0cur  = (t & 1) ? h0b : h0a;
        const float* h0prev = (t == 0) ? nullptr : ((t & 1) ? h0a : h0b);

        // layer 0: x slice at time t has row stride T*IN
        lstm_step_kernel<<<grid, block, 0, stream>>>(
            x + (long)t * IN_DIM, T_STEPS * IN_DIM, IN_DIM, Wx0,
            h0prev, H_DIM, Wh0, b0, Wc0, c0, h0cur, H_DIM);

        // layer 1: input = h0cur; previous h read directly from d_out[:, t-1, :]
        const float* h1prev = (t == 0) ? nullptr : (out + (long)(t - 1) * H_DIM);
        lstm_step_kernel<<<grid, block, 0, stream>>>(
            h0cur, H_DIM, H_DIM, Wx1,
            h1prev, T_STEPS * H_DIM, Wh1, b1, Wc1, c1,
            out + (long)t * H_DIM, T_STEPS * H_DIM);
    }

    finalize_kernel<<<(N_BATCH * H_DIM + 255) / 256, 256, 0, stream>>>(c1, out);
}